// T_GCN2_7327214207529
// MI455X (gfx1250) — compile-verified
//
#include <hip/hip_runtime.h>
#include <hip/hip_bf16.h>
#include <math.h>

// ---------------------------------------------------------------------------
// T-GCN (single step, H0=0) + edge link predictor for MI455X (gfx1250).
// f32 end-to-end; matrix ops via V_WMMA_F32_16X16X4_F32 (wave32).
// ---------------------------------------------------------------------------

typedef float v2f __attribute__((ext_vector_type(2)));
typedef float v8f __attribute__((ext_vector_type(8)));

#define DIN 128
#define HID 32

__device__ __forceinline__ float sigmoidf_(float x) {
    return 1.0f / (1.0f + __expf(-x));
}

__device__ __forceinline__ long long clampidx(long long v, long long n) {
    return v < 0 ? 0 : (v >= n ? n - 1 : v);
}

// --- deg[i] = 1.0 (self loop) --------------------------------------------
__global__ void k_init_deg(float* deg, int n) {
    int i = blockIdx.x * blockDim.x + threadIdx.x;
    if (i < n) deg[i] = 1.0f;
}

// --- deg[dst[e]] += 1 ----------------------------------------------------
__global__ void k_deg_accum(const long long* dst, float* deg, long long E, long long n) {
    long long e = (long long)blockIdx.x * blockDim.x + threadIdx.x;
    if (e >= E) return;
    long long d = clampidx(dst[e], n);
    atomicAdd(&deg[d], 1.0f);
}

// --- deg -> rsqrt(deg) in place ------------------------------------------
__global__ void k_rsqrt(float* deg, int n) {
    int i = blockIdx.x * blockDim.x + threadIdx.x;
    if (i < n) deg[i] = rsqrtf(deg[i]);
}

// --- ax[i,:] = x[i,:] * dinv[i]^2  (self-loop term, no atomics) ----------
__global__ void k_init_ax(const float* __restrict__ x, const float* __restrict__ dinv,
                          float* __restrict__ ax, long long n) {
    long long gid = (long long)blockIdx.x * blockDim.x + threadIdx.x;
    long long i = gid >> 5;          // node
    int q = (int)(gid & 31);         // float4 chunk (32 * 4 = 128)
    if (i >= n) return;
    float s = dinv[i]; s *= s;
    float4 v = ((const float4*)x)[i * 32 + q];
    v.x *= s; v.y *= s; v.z *= s; v.w *= s;
    ((float4*)ax)[i * 32 + q] = v;
}

// --- ax[dst] += x[src] * dinv[src]*dinv[dst]  (one wave per edge) --------
__global__ void k_edge_agg(const long long* __restrict__ src, const long long* __restrict__ dst,
                           const float* __restrict__ x, const float* __restrict__ dinv,
                           float* __restrict__ ax, long long E, long long n) {
    long long gid = (long long)blockIdx.x * blockDim.x + threadIdx.x;
    long long e = gid >> 5;
    int lane = threadIdx.x & 31;
    if (e >= E) return;
    long long s = clampidx(src[e], n);
    long long d = clampidx(dst[e], n);
    float nrm = dinv[s] * dinv[d];
    float4 v = ((const float4*)(x + s * DIN))[lane];
    float* o = ax + d * DIN + lane * 4;
    atomicAdd(o + 0, v.x * nrm);
    atomicAdd(o + 1, v.y * nrm);
    atomicAdd(o + 2, v.z * nrm);
    atomicAdd(o + 3, v.w * nrm);
}

// --- Fold GRU weights: Weff_g = W_g @ L_g[:32,:], beff_g = b_g@L_g[:32] + Lb_g.
//     Written directly in WMMA B-fragment order:
//     bfrag[(((g*32+kk)*2+nt)*64) + lane*2 + j] = Weff[kk*4 + (lane>>4)*2 + j][nt*16 + (lane&15)]
__global__ void k_fuse_weights(const float* __restrict__ Wz, const float* __restrict__ bz,
                               const float* __restrict__ Lz, const float* __restrict__ Lzb,
                               const float* __restrict__ Wh, const float* __restrict__ bh,
                               const float* __restrict__ Lh, const float* __restrict__ Lhb,
                               float* __restrict__ bfrag, float* __restrict__ beff) {
    int t = threadIdx.x;
    for (int g = 0; g < 2; ++g) {
        const float* W = g ? Wh : Wz;
        const float* L = g ? Lh : Lz;   // [64,32] row-major; use top 32 rows
        for (int idx = t; idx < DIN * HID; idx += blockDim.x) {
            int k = idx >> 5, nn = idx & 31;
            float acc = 0.0f;
            for (int j = 0; j < HID; ++j) acc += W[k * HID + j] * L[j * HID + nn];
            int kk = k >> 2, klo = k & 3;
            int lane = (klo >> 1) * 16 + (nn & 15);
            int nt = nn >> 4, jj = klo & 1;
            bfrag[(((g * 32 + kk) * 2 + nt) * 64) + lane * 2 + jj] = acc;
        }
    }
    if (t < 64) {
        int g = t >> 5, nn = t & 31;
        const float* b  = g ? bh  : bz;
        const float* L  = g ? Lh  : Lz;
        const float* Lb = g ? Lhb : Lzb;
        float acc = Lb[nn];
        for (int j = 0; j < HID; ++j) acc += b[j] * L[j * HID + nn];
        beff[g * 32 + nn] = acc;
    }
}

// --- Node GEMM + GRU epilogue:  H = (1 - sigmoid(ax@Wz+bz)) * tanh(ax@Wh+bh)
//     One wave computes a 16x32 tile of H.  128 x v_wmma_f32_16x16x4_f32 per wave.
__global__ void k_node_gemm(const float* __restrict__ ax, const float* __restrict__ bfrag,
                            const float* __restrict__ beff, float* __restrict__ H, int n) {
    int lane = threadIdx.x & 31;
    int wv   = threadIdx.x >> 5;
    int tile = blockIdx.x * 8 + wv;
    int tiles = (n + 15) >> 4;
    if (tile >= tiles) return;

    int mlo   = lane & 15;          // A row within tile / C column
    int khalf = (lane >> 4) * 2;    // A K-subcolumn base for this lane half
    int row = tile * 16 + mlo; if (row >= n) row = n - 1;
    const float* arow = ax + (size_t)row * DIN + khalf;

    v8f accZ0 = {}; v8f accZ1 = {}; v8f accH0 = {}; v8f accH1 = {};
    for (int kk = 0; kk < 32; ++kk) {
        v2f a = *(const v2f*)(arow + kk * 4);
        const float* bz = bfrag + (kk * 2) * 64 + lane * 2;          // gate Z
        const float* bh = bfrag + ((32 + kk) * 2) * 64 + lane * 2;   // gate H
        v2f bz0 = *(const v2f*)(bz);
        v2f bz1 = *(const v2f*)(bz + 64);
        v2f bh0 = *(const v2f*)(bh);
        v2f bh1 = *(const v2f*)(bh + 64);
        accZ0 = __builtin_amdgcn_wmma_f32_16x16x4_f32(false, a, false, bz0, (short)0, accZ0, false, false);
        accZ1 = __builtin_amdgcn_wmma_f32_16x16x4_f32(false, a, false, bz1, (short)0, accZ1, false, false);
        accH0 = __builtin_amdgcn_wmma_f32_16x16x4_f32(false, a, false, bh0, (short)0, accH0, false, false);
        accH1 = __builtin_amdgcn_wmma_f32_16x16x4_f32(false, a, false, bh1, (short)0, accH1, false, false);
    }

    float bzlo = beff[mlo], bzhi = beff[16 + mlo];
    float bhlo = beff[32 + mlo], bhhi = beff[48 + mlo];
    #pragma unroll
    for (int r = 0; r < 8; ++r) {
        int m = (lane < 16) ? r : (r + 8);       // C/D layout: VGPR r -> M=r / M=r+8
        int orow = tile * 16 + m;
        if (orow < n) {
            float z0 = sigmoidf_(accZ0[r] + bzlo);
            float t0 = tanhf(accH0[r] + bhlo);
            H[(size_t)orow * HID + mlo] = (1.0f - z0) * t0;
            float z1 = sigmoidf_(accZ1[r] + bzhi);
            float t1 = tanhf(accH1[r] + bhhi);
            H[(size_t)orow * HID + 16 + mlo] = (1.0f - z1) * t1;
        }
    }
}

// --- Edge MLP: out[e] = sigmoid( relu((H[s]*H[d]) @ w1 + b1) @ w2 + b2 )
//     16 edges per wave; layer-1 via WMMA, layer-2 via shfl-xor reduction.
#define EPAD 36
__global__ void k_edge_mlp(const long long* __restrict__ src, const long long* __restrict__ dst,
                           const float* __restrict__ H,
                           const float* __restrict__ w1, const float* __restrict__ b1,
                           const float* __restrict__ w2, const float* __restrict__ b2v,
                           float* __restrict__ out, long long E, long long n) {
    __shared__ float lds[8 * 16 * EPAD];
    int lane = threadIdx.x & 31;
    int wv   = threadIdx.x >> 5;
    long long base = ((long long)blockIdx.x * 8 + wv) * 16;
    if (base >= E) return;
    float* my = lds + wv * 16 * EPAD;

    int m    = lane & 15;     // edge within tile
    int half = lane >> 4;     // feature half (0: f[0..15], 1: f[16..31])
    long long e = base + m; if (e >= E) e = E - 1;
    long long s = clampidx(src[e], n);
    long long d = clampidx(dst[e], n);

    const float4* Hs = (const float4*)(H + s * HID + half * 16);
    const float4* Hd = (const float4*)(H + d * HID + half * 16);
    float4* rowp = (float4*)(my + m * EPAD + half * 16);
    #pragma unroll
    for (int q = 0; q < 4; ++q) {
        float4 a = Hs[q], b = Hd[q];
        float4 p; p.x = a.x * b.x; p.y = a.y * b.y; p.z = a.z * b.z; p.w = a.w * b.w;
        rowp[q] = p;
    }

    // w1 B fragments (held in VGPRs): w1 is [32 in][32 out] row-major.
    int kh2 = half * 2;
    v2f w1f[8][2];
    #pragma unroll
    for (int kk = 0; kk < 8; ++kk) {
        #pragma unroll
        for (int nt = 0; nt < 2; ++nt) {
            int kb = kk * 4 + kh2;
            int nn = nt * 16 + m;
            v2f f; f[0] = w1[kb * HID + nn]; f[1] = w1[(kb + 1) * HID + nn];
            w1f[kk][nt] = f;
        }
    }

    v8f acc0 = {}; v8f acc1 = {};
    #pragma unroll
    for (int kk = 0; kk < 8; ++kk) {
        v2f a = *(const v2f*)(my + m * EPAD + kk * 4 + kh2);  // per-wave LDS, in-order DS pipe
        acc0 = __builtin_amdgcn_wmma_f32_16x16x4_f32(false, a, false, w1f[kk][0], (short)0, acc0, false, false);
        acc1 = __builtin_amdgcn_wmma_f32_16x16x4_f32(false, a, false, w1f[kk][1], (short)0, acc1, false, false);
    }

    float b1lo = b1[m], b1hi = b1[m + 16];
    float w2lo = w2[m], w2hi = w2[m + 16];
    float bias2 = b2v[0];
    #pragma unroll
    for (int r = 0; r < 8; ++r) {
        float p = fmaxf(acc0[r] + b1lo, 0.0f) * w2lo + fmaxf(acc1[r] + b1hi, 0.0f) * w2hi;
        p += __shfl_xor(p, 1);
        p += __shfl_xor(p, 2);
        p += __shfl_xor(p, 4);
        p += __shfl_xor(p, 8);   // reduce within each 16-lane half
        if (m == 0) {
            long long eo = base + r + half * 8;   // lane0 -> M=r, lane16 -> M=r+8
            if (eo < E) out[eo] = sigmoidf_(p + bias2);
        }
    }
}

// ---------------------------------------------------------------------------
extern "C" void kernel_launch(void* const* d_in, const int* in_sizes, int n_in,
                              void* d_out, int out_size, void* d_ws, size_t ws_size,
                              hipStream_t stream) {
    const float*     x    = (const float*)d_in[0];
    const long long* ei   = (const long long*)d_in[1];   // int64 per reference
    const float*     W_z  = (const float*)d_in[2];
    const float*     b_z  = (const float*)d_in[3];
    // d_in[4], d_in[5] = W_r, b_r : dead (R*H0 == 0)
    const float*     W_h  = (const float*)d_in[6];
    const float*     b_h  = (const float*)d_in[7];
    const float*     Lz_w = (const float*)d_in[8];
    const float*     Lz_b = (const float*)d_in[9];
    // d_in[10], d_in[11] = Lr_w, Lr_b : dead
    const float*     Lh_w = (const float*)d_in[12];
    const float*     Lh_b = (const float*)d_in[13];
    const float*     w1   = (const float*)d_in[14];
    const float*     b1   = (const float*)d_in[15];
    const float*     w2   = (const float*)d_in[16];
    const float*     b2   = (const float*)d_in[17];
    float* out = (float*)d_out;

    const long long N = in_sizes[0] / DIN;
    const long long E = in_sizes[1] / 2;
    const long long* src = ei;
    const long long* dst = ei + E;

    // Workspace carve-up (~64.6 MB): dinv | ax | bfrag | beff | H
    char* w = (char*)d_ws;
    size_t o = 0;
    auto carve = [&](size_t bytes) { char* p = w + o; o = (o + bytes + 255) & ~(size_t)255; return p; };
    float* dinv  = (float*)carve((size_t)N * 4);
    float* ax    = (float*)carve((size_t)N * DIN * 4);
    float* bfrag = (float*)carve((size_t)2 * 32 * 2 * 64 * 4);   // 32 KB
    float* beff  = (float*)carve((size_t)64 * 4);
    float* H     = (float*)carve((size_t)N * HID * 4);
    (void)ws_size; (void)n_in; (void)out_size;

    const int B = 256;

    k_init_deg  <<<(unsigned)((N + B - 1) / B), B, 0, stream>>>(dinv, (int)N);
    k_deg_accum <<<(unsigned)((E + B - 1) / B), B, 0, stream>>>(dst, dinv, E, N);
    k_rsqrt     <<<(unsigned)((N + B - 1) / B), B, 0, stream>>>(dinv, (int)N);
    k_init_ax   <<<(unsigned)((N * 32 + B - 1) / B), B, 0, stream>>>(x, dinv, ax, N);
    k_edge_agg  <<<(unsigned)((E * 32 + B - 1) / B), B, 0, stream>>>(src, dst, x, dinv, ax, E, N);
    k_fuse_weights<<<1, B, 0, stream>>>(W_z, b_z, Lz_w, Lz_b, W_h, b_h, Lh_w, Lh_b, bfrag, beff);

    int tiles = (int)((N + 15) / 16);
    k_node_gemm <<<(unsigned)((tiles + 7) / 8), B, 0, stream>>>(ax, bfrag, beff, H, (int)N);
    k_edge_mlp  <<<(unsigned)((E + 127) / 128), B, 0, stream>>>(src, dst, H, w1, b1, w2, b2, out, E, N);
}